// GraphClassifier_84825604096601
// MI455X (gfx1250) — compile-verified
//
#include <hip/hip_runtime.h>
#include <hip/hip_bf16.h>

typedef __attribute__((ext_vector_type(2))) float v2f;
typedef __attribute__((ext_vector_type(8))) float v8f;

#define N_NODES   100000
#define N_EDGES   3200000
#define N_FEAT    512
#define HID1      128
#define HID2      16
#define N_CLASSES 2

// ---------------------------------------------------------------- utilities

__global__ void k_zero(float* __restrict__ p, long long n) {
    long long t = (long long)blockIdx.x * blockDim.x + threadIdx.x;
    if (t < n) p[t] = 0.0f;
}

__global__ void k_fill1(float* __restrict__ p, long long n) {
    long long t = (long long)blockIdx.x * blockDim.x + threadIdx.x;
    if (t < n) p[t] = 1.0f;   // self-loop contributes 1 to every node's degree
}

__global__ void k_deg_accum(const int* __restrict__ dst, float* __restrict__ deg, int n_edges) {
    long long t = (long long)blockIdx.x * blockDim.x + threadIdx.x;
    if (t < n_edges) atomicAdd(&deg[dst[t]], 1.0f);
}

__global__ void k_deg_finalize(float* __restrict__ deg, int n) {
    long long t = (long long)blockIdx.x * blockDim.x + threadIdx.x;
    if (t < n) {
        float d = deg[t];
        deg[t] = rsqrtf(d > 0.0f ? d : 1.0f);   // becomes dinv
    }
}

// ---------------------------------------------------------------- GEMM1: H = X @ W1  (100000x512 @ 512x128), fp32 WMMA
// block = 256 thr = 8 waves. blockIdx.x = M-tile (16 rows). wave w owns N-tile w (16 cols).
// A-tile (16x512 f32 = 32KB) staged into LDS once per block via async-to-LDS;
// all 8 waves then feed WMMA A-fragments from LDS (ds_load_b64).
__global__ void k_gemm1_wmma(const float* __restrict__ X, const float* __restrict__ W,
                             float* __restrict__ H) {
    __shared__ float at[16 * N_FEAT];          // 32 KB A tile

    const int tid  = threadIdx.x;
    const int wave = tid >> 5;
    const int lane = tid & 31;
    const int hh   = lane >> 4;                // half-wave select (K pair)
    const int r    = lane & 15;                // row within A / col within B
    const int m0   = blockIdx.x * 16;
    const int n0   = wave * 16;

    // ---- async stage: 16x512 floats = 2048 B128 transfers = 256 thr x 8 iters
    {
        const float*   gsrc  = X + (size_t)m0 * N_FEAT;
        const unsigned lbase = (unsigned)(unsigned long long)(void*)&at[0];
#pragma unroll
        for (int it = 0; it < 8; ++it) {
            const int      fidx  = (it * 256 + tid) * 4;        // float index (16B granules)
            const unsigned laddr = lbase + (unsigned)fidx * 4u; // LDS byte address
            const float*   gaddr = gsrc + fidx;
            asm volatile("global_load_async_to_lds_b128 %0, %1, off"
                         :: "v"(laddr), "v"(gaddr)
                         : "memory");
        }
        asm volatile("s_wait_asynccnt 0" ::: "memory");
        __syncthreads();
    }

    const float* __restrict__ arow = &at[r * N_FEAT];   // LDS row for A frags
    const float* __restrict__ bcol = W + n0 + r;        // B: col r of tile (global, L2-hit)

    v8f acc = {};
    for (int k = 0; k < N_FEAT; k += 4) {
        v2f a, b;
        // A 16x4 frag: lanes0-15 K={k,k+1}, lanes16-31 K={k+2,k+3} -> contiguous ds_load_b64
        a.x = arow[k + 2 * hh];
        a.y = arow[k + 2 * hh + 1];
        // B 4x16 frag: same K split, column n0+r
        b.x = bcol[(size_t)(k + 2 * hh) * HID1];
        b.y = bcol[(size_t)(k + 2 * hh + 1) * HID1];
        acc = __builtin_amdgcn_wmma_f32_16x16x4_f32(false, a, false, b,
                                                    (short)0, acc, false, false);
    }
    // C/D layout: VGPR g -> row m0 + 8*hh + g, col n0 + r
    float* __restrict__ hp = H + (size_t)(m0 + hh * 8) * HID1 + n0 + r;
#pragma unroll
    for (int g = 0; g < 8; ++g) hp[(size_t)g * HID1] = acc[g];
}

// ---------------------------------------------------------------- GEMM2: H2 = A @ W2  (100000x128 @ 128x16), fp32 WMMA
// 8 waves per block, each wave owns one 16-row M-tile; single N-tile (16 cols).
__global__ void k_gemm2_wmma(const float* __restrict__ A, const float* __restrict__ W,
                             float* __restrict__ H2) {
    const int wave  = threadIdx.x >> 5;
    const int lane  = threadIdx.x & 31;
    const int mtile = blockIdx.x * 8 + wave;
    if (mtile * 16 >= N_NODES) return;   // wave-uniform exit (EXEC stays all-ones)
    const int hh = lane >> 4;
    const int r  = lane & 15;
    const int m0 = mtile * 16;

    const float* __restrict__ arow = A + (size_t)(m0 + r) * HID1;
    const float* __restrict__ bcol = W + r;

    v8f acc = {};
    for (int k = 0; k < HID1; k += 4) {
        v2f a, b;
        a.x = arow[k + 2 * hh];
        a.y = arow[k + 2 * hh + 1];
        b.x = bcol[(size_t)(k + 2 * hh) * HID2];
        b.y = bcol[(size_t)(k + 2 * hh + 1) * HID2];
        acc = __builtin_amdgcn_wmma_f32_16x16x4_f32(false, a, false, b,
                                                    (short)0, acc, false, false);
    }
    float* __restrict__ hp = H2 + (size_t)(m0 + hh * 8) * HID2 + r;
#pragma unroll
    for (int g = 0; g < 8; ++g) hp[(size_t)g * HID2] = acc[g];
}

// ---------------------------------------------------------------- edge scatter, layer 1 (128 feats): one wave per edge
__global__ void k_scatter1(const float* __restrict__ H, const int* __restrict__ src,
                           const int* __restrict__ dst, const float* __restrict__ dinv,
                           float* __restrict__ AGG, int n_edges) {
    long long t = (long long)blockIdx.x * blockDim.x + threadIdx.x;
    int e = (int)(t >> 5);
    if (e >= n_edges) return;
    const int lane = threadIdx.x & 31;
    const int s = src[e], d = dst[e];        // wave-uniform scalar loads
    const float w = dinv[s] * dinv[d];
    const float* __restrict__ hs = H + (size_t)s * HID1;
    float* __restrict__ od = AGG + (size_t)d * HID1;
#pragma unroll
    for (int j = 0; j < 4; ++j) {
        int f = lane + 32 * j;
        atomicAdd(&od[f], w * hs[f]);
    }
}

// fused self-loop + bias + ReLU:  AGG = relu(AGG + dinv[i]^2 * H + b1)
__global__ void k_self_bias_relu1(float* __restrict__ AGG, const float* __restrict__ H,
                                  const float* __restrict__ dinv, const float* __restrict__ b1) {
    long long t = (long long)blockIdx.x * blockDim.x + threadIdx.x;
    if (t >= (long long)N_NODES * HID1) return;
    int i = (int)(t >> 7);
    int f = (int)(t & (HID1 - 1));
    float di = dinv[i];
    float v = AGG[t] + di * di * H[t] + b1[f];
    AGG[t] = v > 0.0f ? v : 0.0f;
}

// ---------------------------------------------------------------- edge scatter, layer 2 (16 feats): one thread per (edge,feat)
__global__ void k_scatter2(const float* __restrict__ H2, const int* __restrict__ src,
                           const int* __restrict__ dst, const float* __restrict__ dinv,
                           float* __restrict__ XE, int n_edges) {
    long long t = (long long)blockIdx.x * blockDim.x + threadIdx.x;
    if (t >= (long long)n_edges * HID2) return;
    int e = (int)(t >> 4);
    int f = (int)(t & (HID2 - 1));
    int s = src[e], d = dst[e];
    float w = dinv[s] * dinv[d];
    atomicAdd(&XE[(size_t)d * HID2 + f], w * H2[(size_t)s * HID2 + f]);
}

__global__ void k_self_bias2(float* __restrict__ XE, const float* __restrict__ H2,
                             const float* __restrict__ dinv, const float* __restrict__ b2) {
    long long t = (long long)blockIdx.x * blockDim.x + threadIdx.x;
    if (t >= (long long)N_NODES * HID2) return;
    int i = (int)(t >> 4);
    int f = (int)(t & (HID2 - 1));
    float di = dinv[i];
    XE[t] = XE[t] + di * di * H2[t] + b2[f];
}

// ---------------------------------------------------------------- classifier: sigmoid(x_emb @ fc_w + fc_b), one thread/node
__global__ void k_classifier(const float* __restrict__ XE, const float* __restrict__ fcw,
                             const float* __restrict__ fcb, float* __restrict__ P) {
    long long i = (long long)blockIdx.x * blockDim.x + threadIdx.x;
    if (i >= N_NODES) return;
    const float* __restrict__ x = XE + (size_t)i * HID2;
#pragma unroll
    for (int c = 0; c < N_CLASSES; ++c) {
        float acc = fcb[c];
#pragma unroll
        for (int k = 0; k < HID2; ++k) acc += x[k] * fcw[k * N_CLASSES + c];
        P[(size_t)i * N_CLASSES + c] = 1.0f / (1.0f + __expf(-acc));
    }
}

// ---------------------------------------------------------------- launch

extern "C" void kernel_launch(void* const* d_in, const int* in_sizes, int n_in,
                              void* d_out, int out_size, void* d_ws, size_t ws_size,
                              hipStream_t stream) {
    const float* X   = (const float*)d_in[0];
    const int*   ei  = (const int*)d_in[1];      // (2, E) row-major int32
    const float* W1  = (const float*)d_in[2];
    const float* b1  = (const float*)d_in[3];
    const float* W2  = (const float*)d_in[4];
    const float* b2  = (const float*)d_in[5];
    const float* fcw = (const float*)d_in[6];
    const float* fcb = (const float*)d_in[7];

    const int* src = ei;
    const int* dst = ei + N_EDGES;

    float* out  = (float*)d_out;
    float* xemb = out;                               // [N_NODES, 16]
    float* prob = out + (size_t)N_NODES * HID2;      // [N_NODES, 2]

    float* ws   = (float*)d_ws;
    float* dinv = ws;                                        // [100000] (deg -> dinv)
    float* H    = ws + (1 << 17);                            // [N, 128]
    float* AGG  = H + (size_t)N_NODES * HID1;                // [N, 128]
    float* H2   = AGG + (size_t)N_NODES * HID1;              // [N, 16]

    const int T = 256;
    auto nb = [](long long n, int t) { return (unsigned)((n + t - 1) / t); };

    // degree / normalization
    k_fill1<<<nb(N_NODES, T), T, 0, stream>>>(dinv, N_NODES);
    k_deg_accum<<<nb(N_EDGES, T), T, 0, stream>>>(dst, dinv, N_EDGES);
    k_deg_finalize<<<nb(N_NODES, T), T, 0, stream>>>(dinv, N_NODES);

    // layer 1
    k_gemm1_wmma<<<N_NODES / 16, T, 0, stream>>>(X, W1, H);
    k_zero<<<nb((long long)N_NODES * HID1, T), T, 0, stream>>>(AGG, (long long)N_NODES * HID1);
    k_scatter1<<<nb((long long)N_EDGES * 32, T), T, 0, stream>>>(H, src, dst, dinv, AGG, N_EDGES);
    k_self_bias_relu1<<<nb((long long)N_NODES * HID1, T), T, 0, stream>>>(AGG, H, dinv, b1);

    // layer 2
    k_gemm2_wmma<<<nb(N_NODES / 16, 8), T, 0, stream>>>(AGG, W2, H2);
    k_zero<<<nb((long long)N_NODES * HID2, T), T, 0, stream>>>(xemb, (long long)N_NODES * HID2);
    k_scatter2<<<nb((long long)N_EDGES * HID2, T), T, 0, stream>>>(H2, src, dst, dinv, xemb, N_EDGES);
    k_self_bias2<<<nb((long long)N_NODES * HID2, T), T, 0, stream>>>(xemb, H2, dinv, b2);

    // classifier
    k_classifier<<<nb(N_NODES, T), T, 0, stream>>>(xemb, fcw, fcb, prob);
}